// PointerNER_52888227283426
// MI455X (gfx1250) — compile-verified
//
#include <hip/hip_runtime.h>
#include <hip/hip_bf16.h>
#include <math.h>

// Problem constants (from reference)
#define SN 65536
#define HN 768
#define TN 6
#define KN 3
#define WINN 15
#define NT 16          // padded N for WMMA tile: [W_start(6) | W_end(6) | zero(4)]

typedef float v2f __attribute__((ext_vector_type(2)));
typedef float v8f __attribute__((ext_vector_type(8)));

// ---------------------------------------------------------------------------
// Pass 1: fused  scores = E @ [Ws|We] + b   and   colsum(E)
// One wave = one 16-row tile. WMMA f32 16x16x4, K-loop of 192.
// A-fragment layout (32-bit A 16x4): lanes 0-15 -> K=0,1 ; lanes 16-31 -> K=2,3
// B-fragment assumed mirror layout:  lane&15 = N ; lane>>4 selects K pair.
// C/D layout: lane&15 = N ; VGPR v = M=v (lanes>=16: M=v+8).
// ---------------------------------------------------------------------------
__global__ __launch_bounds__(256) void pass1_scores_colsum(
    const float* __restrict__ E,
    const float* __restrict__ Ws, const float* __restrict__ bs,
    const float* __restrict__ We, const float* __restrict__ be,
    float* __restrict__ scS,   // [T][S]
    float* __restrict__ scE,   // [T][S]
    float* __restrict__ colsum // [H], pre-zeroed
)
{
    __shared__ v2f   Bpack[(HN / 2) * NT]; // 6144 float2 = 48 KiB
    __shared__ float lds_cs[HN];           // 3 KiB

    const int tid = threadIdx.x;

    // Stage padded B operand: Bpack[kk*16+n] = { Wcat[2kk][n], Wcat[2kk+1][n] }
    for (int i = tid; i < (HN / 2) * NT; i += 256) {
        const int kk = i >> 4;
        const int n  = i & 15;
        const int h0 = kk * 2;
        float x0 = 0.0f, x1 = 0.0f;
        if (n < TN) {
            x0 = Ws[h0 * TN + n];
            x1 = Ws[(h0 + 1) * TN + n];
        } else if (n < 2 * TN) {
            x0 = We[h0 * TN + (n - TN)];
            x1 = We[(h0 + 1) * TN + (n - TN)];
        }
        v2f b; b.x = x0; b.y = x1;
        Bpack[i] = b;
    }
    for (int c = tid; c < HN; c += 256) lds_cs[c] = 0.0f;
    __syncthreads();

    const int wave     = tid >> 5;
    const int lane     = tid & 31;
    const int m        = lane & 15;   // row within tile (A role) / column N (B,C role)
    const int khalf    = lane >> 4;
    const int row_base = (blockIdx.x * 8 + wave) * 16;

    const float* Arow = E + (size_t)(row_base + m) * HN + khalf * 2;

    v8f acc = {};
#pragma unroll 4
    for (int kb = 0; kb < HN / 4; ++kb) {
        v2f a = *(const v2f*)(Arow + kb * 4);
        v2f b = Bpack[(kb * 2 + khalf) * NT + m];
        acc = __builtin_amdgcn_wmma_f32_16x16x4_f32(
            /*neg_a=*/false, a, /*neg_b=*/false, b,
            /*c_mod=*/(short)0, acc, /*reuse_a=*/false, /*reuse_b=*/false);
    }

    // Bias + score store (only N columns 0..11 are live)
    const int n = m;
    if (n < 2 * TN) {
        const float bias  = (n < TN) ? bs[n] : be[n - TN];
        float* plane      = (n < TN) ? (scS + (size_t)n * SN)
                                     : (scE + (size_t)(n - TN) * SN);
        const int mofs = khalf * 8;
#pragma unroll
        for (int v = 0; v < 8; ++v)
            plane[row_base + mofs + v] = acc[v] + bias;
    }

    // Fused column-sum for pooled mean: coalesced re-read of the hot tile.
    const float* tile = E + (size_t)row_base * HN;
#pragma unroll
    for (int j = 0; j < HN / 32; ++j) {
        const int c = j * 32 + lane;
        float s = 0.0f;
#pragma unroll 4
        for (int r = 0; r < 16; ++r) s += tile[(size_t)r * HN + c];
        atomicAdd(&lds_cs[c], s);     // ds_add_f32, bank-conflict free
    }
    __syncthreads();
    for (int c = tid; c < HN; c += 256) atomicAdd(&colsum[c], lds_cs[c]);
}

// ---------------------------------------------------------------------------
// Pass 2a: per-type (6 blocks) softmax stats + top-3 over S=65536 scores.
// stats[t*8]: 0=max 1=sumexp 2..4=top3 probs 5..7=top3 idx (int bits)
// ---------------------------------------------------------------------------
__global__ __launch_bounds__(256) void pass2a_stats(
    const float* __restrict__ scS, float* __restrict__ stats)
{
    const int t   = blockIdx.x;
    const int tid = threadIdx.x;
    const float* sc = scS + (size_t)t * SN;

    __shared__ float red[256];
    __shared__ float tv[256 * 3];
    __shared__ int   ti[256 * 3];

    float mx = -INFINITY;
    float v0 = -INFINITY, v1 = -INFINITY, v2 = -INFINITY;
    int   i0 = 0, i1 = 0, i2 = 0;
    for (int i = tid; i < SN; i += 256) {
        const float v = sc[i];
        mx = fmaxf(mx, v);
        if (v > v2) {                 // strict > keeps lowest index on ties
            if (v > v0)      { v2 = v1; i2 = i1; v1 = v0; i1 = i0; v0 = v; i0 = i; }
            else if (v > v1) { v2 = v1; i2 = i1; v1 = v;  i1 = i; }
            else             { v2 = v;  i2 = i; }
        }
    }
    red[tid] = mx; __syncthreads();
    for (int s = 128; s > 0; s >>= 1) {
        if (tid < s) red[tid] = fmaxf(red[tid], red[tid + s]);
        __syncthreads();
    }
    const float gmax = red[0]; __syncthreads();

    float se = 0.0f;
    for (int i = tid; i < SN; i += 256) se += expf(sc[i] - gmax);
    red[tid] = se; __syncthreads();
    for (int s = 128; s > 0; s >>= 1) {
        if (tid < s) red[tid] += red[tid + s];
        __syncthreads();
    }
    const float gsum = red[0];

    tv[tid * 3 + 0] = v0; tv[tid * 3 + 1] = v1; tv[tid * 3 + 2] = v2;
    ti[tid * 3 + 0] = i0; ti[tid * 3 + 1] = i1; ti[tid * 3 + 2] = i2;
    __syncthreads();

    if (tid == 0) {
        float b0 = -INFINITY, b1 = -INFINITY, b2 = -INFINITY;
        int   j0 = 0, j1 = 0, j2 = 0;
        for (int e = 0; e < 256 * 3; ++e) {
            const float v = tv[e]; const int idx = ti[e];
            if ((v > b2) || (v == b2 && idx < j2)) {
                if ((v > b0) || (v == b0 && idx < j0)) {
                    b2 = b1; j2 = j1; b1 = b0; j1 = j0; b0 = v; j0 = idx;
                } else if ((v > b1) || (v == b1 && idx < j1)) {
                    b2 = b1; j2 = j1; b1 = v; j1 = idx;
                } else { b2 = v; j2 = idx; }
            }
        }
        float* st = stats + t * 8;
        st[0] = gmax; st[1] = gsum;
        st[2] = expf(b0 - gmax) / gsum;
        st[3] = expf(b1 - gmax) / gsum;
        st[4] = expf(b2 - gmax) / gsum;
        ((int*)st)[5] = j0; ((int*)st)[6] = j1; ((int*)st)[7] = j2;
    }
}

// ---------------------------------------------------------------------------
// Pass 2b: pooled MLP: h = gelu(mean(E)@W1 + b1); type_conf = sigmoid(h@W2+b2)
// ---------------------------------------------------------------------------
__global__ __launch_bounds__(64) void pass2b_mlp(
    const float* __restrict__ colsum,
    const float* __restrict__ W1, const float* __restrict__ b1,
    const float* __restrict__ W2, const float* __restrict__ b2,
    float* __restrict__ type_conf)
{
    __shared__ float h[64];
    const int j = threadIdx.x;
    float a = b1[j];
    const float inv_s = 1.0f / (float)SN;
    for (int c = 0; c < HN; ++c)
        a += (colsum[c] * inv_s) * W1[c * 64 + j];
    h[j] = 0.5f * a * (1.0f + erff(a * 0.70710678118654752f)); // exact gelu
    __syncthreads();
    if (j < TN) {
        float z = b2[j];
        for (int q = 0; q < 64; ++q) z += h[q] * W2[q * TN + j];
        type_conf[j] = 1.0f / (1.0f + expf(-z));
    }
}

// ---------------------------------------------------------------------------
// Pass 3: end-window argmax + gating, write (conf, starts, ends, valid) = 72 f32
// ---------------------------------------------------------------------------
__global__ __launch_bounds__(32) void pass3_finalize(
    const float* __restrict__ scE, const float* __restrict__ stats,
    const float* __restrict__ type_conf, float* __restrict__ out)
{
    const int id = threadIdx.x;
    if (id >= TN * KN) return;
    const int t = id / KN;
    const float* st = stats + t * 8;
    const float pval = st[2 + (id % KN)];
    const int   sidx = ((const int*)st)[5 + (id % KN)];

    const float* ep = scE + (size_t)t * SN;
    float best = -INFINITY; int boff = 0;
#pragma unroll
    for (int w = 0; w < WINN; ++w) {
        const int pos = sidx + w;
        const float v = (pos < SN) ? ep[pos] : -INFINITY;
        if (v > best) { best = v; boff = w; }   // first max wins (argmax semantics)
    }
    const float tc   = type_conf[t];
    const bool valid = (tc >= 0.3f) && (pval >= 0.15f);

    out[0 * (TN * KN) + id] = valid ? pval * tc : 0.0f;
    out[1 * (TN * KN) + id] = (float)sidx;
    out[2 * (TN * KN) + id] = (float)(sidx + boff + 1);
    out[3 * (TN * KN) + id] = valid ? 1.0f : 0.0f;
}

// ---------------------------------------------------------------------------
extern "C" void kernel_launch(void* const* d_in, const int* in_sizes, int n_in,
                              void* d_out, int out_size, void* d_ws, size_t ws_size,
                              hipStream_t stream)
{
    const float* E  = (const float*)d_in[0];
    const float* Ws = (const float*)d_in[1];
    const float* bs = (const float*)d_in[2];
    const float* We = (const float*)d_in[3];
    const float* be = (const float*)d_in[4];
    const float* W1 = (const float*)d_in[5];
    const float* b1 = (const float*)d_in[6];
    const float* W2 = (const float*)d_in[7];
    const float* b2 = (const float*)d_in[8];

    float* ws     = (float*)d_ws;
    float* scS    = ws;                                  // 6*65536
    float* scE    = scS + (size_t)TN * SN;               // 6*65536
    float* colsum = scE + (size_t)TN * SN;               // 768
    float* stats  = colsum + HN;                         // 6*8
    float* tconf  = stats + TN * 8;                      // 6

    hipMemsetAsync(colsum, 0, HN * sizeof(float), stream);

    pass1_scores_colsum<<<SN / 128, 256, 0, stream>>>(E, Ws, bs, We, be,
                                                      scS, scE, colsum);
    pass2a_stats<<<TN, 256, 0, stream>>>(scS, stats);
    pass2b_mlp<<<1, 64, 0, stream>>>(colsum, W1, b1, W2, b2, tconf);
    pass3_finalize<<<1, 32, 0, stream>>>(scE, stats, tconf, (float*)d_out);
}